// GNN_27187142983848
// MI455X (gfx1250) — compile-verified
//
#include <hip/hip_runtime.h>
#include <stdint.h>

#define N_NODES 100000
#define N_EDGES 800000
#define HID 32
#define HEADS 4
#define HC 128
#define EDIM 16
#define G_GRAPHS 256
#define OUT_C 2
#define SLOPE 0.2f
#define BN_EPS 1e-5f

typedef __attribute__((ext_vector_type(2))) float v2f;
typedef __attribute__((ext_vector_type(8))) float v8f;

// monotone order-preserving float<->uint encoding for atomicMax
static __device__ __forceinline__ unsigned encf(float x) {
  unsigned u = __float_as_uint(x);
  return (u & 0x80000000u) ? ~u : (u | 0x80000000u);
}
static __device__ __forceinline__ float decf(unsigned u) {
  return (u & 0x80000000u) ? __uint_as_float(u ^ 0x80000000u) : __uint_as_float(~u);
}

// ---- weight prep: Ws[l][k][n] (f32) -> WT[l][n][k] (f32, transposed) ----
__global__ void k_prep_w(const float* __restrict__ Ws, float* __restrict__ WT) {
  int t = blockIdx.x * blockDim.x + threadIdx.x;
  if (t >= 3 * 128 * 128) return;
  int l = t / 16384, r = t % 16384, k = r / 128, n = r % 128;
  WT[l * 16384 + n * 128 + k] = Ws[t];
}

// ---- fold lin_e with att_edge: weff[l][h][j] = sum_c lin_e[l][j][h*32+c]*att_edge[l][h][c] ----
__global__ void k_weff(const float* __restrict__ lin_e, const float* __restrict__ att_edge,
                       float* __restrict__ weff) {
  int t = blockIdx.x * blockDim.x + threadIdx.x;
  if (t >= 3 * 128) return;
  int l = t / 128, r = t % 128, hh = r / 32, j = r % 32;
  float s = 0.f;
  #pragma unroll
  for (int c = 0; c < 32; ++c)
    s += lin_e[l * 32 * 128 + j * 128 + hh * 32 + c] * att_edge[l * 128 + hh * 32 + c];
  weff[t] = s;
}

// ---- per-edge: e_h = edge_attr @ edge_fc_W + b; alpha_edge[l][e][h] = e_h · weff[l][h] ----
__global__ void k_edge_feat(const float* __restrict__ eattr, const float* __restrict__ Wfc,
                            const float* __restrict__ bfc, const float* __restrict__ weff,
                            float* __restrict__ aeall) {
  int e = blockIdx.x * blockDim.x + threadIdx.x;
  if (e >= N_EDGES) return;
  float ea[EDIM];
  #pragma unroll
  for (int k = 0; k < EDIM; ++k) ea[k] = eattr[(size_t)e * EDIM + k];
  float eh[HID];
  #pragma unroll
  for (int c = 0; c < HID; ++c) {
    float s = bfc[c];
    #pragma unroll
    for (int k = 0; k < EDIM; ++k) s += ea[k] * Wfc[k * HID + c];
    eh[c] = s;
  }
  for (int l = 0; l < 3; ++l)
    for (int hh = 0; hh < HEADS; ++hh) {
      const float* w = weff + l * 128 + hh * 32;
      float s = 0.f;
      #pragma unroll
      for (int j = 0; j < HID; ++j) s += eh[j] * w[j];
      aeall[((size_t)l * N_EDGES + e) * HEADS + hh] = s;
    }
}

// ---- WMMA GEMM (exact f32): xl[M,128] = h[M,128] @ W[128,128] (WT transposed) ----
// V_WMMA_F32_16X16X4_F32 fragments: A(16x4) and B(4x16) each hold 2 consecutive
// K-elements per lane: K {0,1} for lanes 0-15, K {2,3} for lanes 16-31.
__global__ __launch_bounds__(256) void k_gemm(const float* __restrict__ h,
                                              const float* __restrict__ WT,
                                              float* __restrict__ xl) {
  const int lane = threadIdx.x & 31;
  const int wv   = threadIdx.x >> 5;        // 8 waves, one 16-wide N-tile each
  const int m0   = blockIdx.x * 16;
  const int n0   = wv * 16;
  const int r    = m0 + (lane & 15);        // A row for this lane
  const int cB   = n0 + (lane & 15);        // B column for this lane
  const int ksub = (lane >= 16) ? 2 : 0;
  const float* arow = h  + (size_t)r  * 128 + ksub;
  const float* bcol = WT + (size_t)cB * 128 + ksub;
  v8f acc = {};
  #pragma unroll
  for (int kk = 0; kk < 32; ++kk) {
    v2f a2 = *(const v2f*)(arow + kk * 4);
    v2f b2 = *(const v2f*)(bcol + kk * 4);
    acc = __builtin_amdgcn_wmma_f32_16x16x4_f32(false, a2, false, b2,
                                                (short)0, acc, false, false);
  }
  #pragma unroll
  for (int j = 0; j < 8; ++j) {
    int row = m0 + j + ((lane >= 16) ? 8 : 0);
    xl[(size_t)row * 128 + cB] = acc[j];
  }
}

// ---- node attention coefficients as/ad [N,H] ----
__global__ void k_attn_node(const float* __restrict__ xl, const float* __restrict__ a_s,
                            const float* __restrict__ a_d, float* __restrict__ asb,
                            float* __restrict__ adb) {
  int t = blockIdx.x * blockDim.x + threadIdx.x;
  if (t >= N_NODES * HEADS) return;
  int n = t >> 2, hh = t & 3;
  const float* row = xl + (size_t)n * 128 + hh * 32;
  const float* ws  = a_s + hh * 32;
  const float* wd  = a_d + hh * 32;
  float s1 = 0.f, s2 = 0.f;
  #pragma unroll
  for (int c = 0; c < 32; ++c) { float v = row[c]; s1 += v * ws[c]; s2 += v * wd[c]; }
  asb[t] = s1; adb[t] = s2;
}

// ---- edge pass A: logits + leaky_relu + segment max (encoded atomicMax) ----
__global__ void k_edge_a(const int* __restrict__ src, const int* __restrict__ dst,
                         const float* __restrict__ asb, const float* __restrict__ adb,
                         const float* __restrict__ ae, float* __restrict__ abuf,
                         unsigned* __restrict__ amax) {
  int t = blockIdx.x * blockDim.x + threadIdx.x;
  if (t >= N_EDGES * HEADS) return;
  int e = t >> 2, hh = t & 3;
  int s = src[e], d = dst[e];
  float a = asb[s * HEADS + hh] + adb[d * HEADS + hh] + ae[t];
  a = a > 0.f ? a : SLOPE * a;
  abuf[t] = a;
  atomicMax(&amax[d * HEADS + hh], encf(a));
}

// ---- edge pass B: e^(a-amax); scatter numerator and denominator ----
__global__ void k_edge_b(const int* __restrict__ src, const int* __restrict__ dst,
                         const float* __restrict__ abuf, const unsigned* __restrict__ amax,
                         const float* __restrict__ xl, float* __restrict__ hnx,
                         float* __restrict__ denom) {
  int t = blockIdx.x * blockDim.x + threadIdx.x;   // E*128 threads exactly
  int e = t >> 7, c = t & 127, hh = c >> 5;
  int s = src[e], d = dst[e];
  float w = __expf(abuf[e * HEADS + hh] - decf(amax[d * HEADS + hh]));
  atomicAdd(&hnx[(size_t)d * 128 + c], w * xl[(size_t)s * 128 + c]);
  if ((c & 31) == 0) atomicAdd(&denom[d * HEADS + hh], w);
}

// ---- divide by denom, add bias, accumulate BN stats (block-partial reduction) ----
__global__ void k_finish_bn(float* __restrict__ hnx, const float* __restrict__ denom,
                            const float* __restrict__ bias, float* __restrict__ bnsum,
                            float* __restrict__ bnsq) {
  int c = threadIdx.x;                     // 128 channel threads
  int hh = c >> 5;
  float b = bias[c];
  float s = 0.f, s2 = 0.f;
  int n0 = blockIdx.x * 64;
  for (int j = 0; j < 64; ++j) {
    int n = n0 + j;
    if (n < N_NODES) {
      float dd = denom[n * HEADS + hh] + 1e-16f;
      float v = hnx[(size_t)n * 128 + c] / dd + b;
      hnx[(size_t)n * 128 + c] = v;
      s += v; s2 += v * v;
    }
  }
  atomicAdd(&bnsum[c], s);
  atomicAdd(&bnsq[c], s2);
}

// ---- BN normalize + ReLU; write next-layer h (f32) ----
__global__ void k_bn_apply(const float* __restrict__ hnx, const float* __restrict__ bnsum,
                           const float* __restrict__ bnsq, const float* __restrict__ g,
                           const float* __restrict__ b, float* __restrict__ hcur) {
  int t = blockIdx.x * blockDim.x + threadIdx.x;
  if (t >= N_NODES * HC) return;
  int c = t & 127;
  const float inv_n = 1.0f / (float)N_NODES;
  float mu  = bnsum[c] * inv_n;
  float var = bnsq[c] * inv_n - mu * mu;
  float v = (hnx[t] - mu) * rsqrtf(var + BN_EPS) * g[c] + b[c];
  hcur[t] = v > 0.f ? v : 0.f;
}

// ---- global mean pool (sum + count) ----
__global__ void k_pool(const float* __restrict__ h, const int* __restrict__ batch,
                       float* __restrict__ pooled, float* __restrict__ cnt) {
  int t = blockIdx.x * blockDim.x + threadIdx.x;
  if (t >= N_NODES * HC) return;
  int n = t >> 7, c = t & 127;
  int bg = batch[n];
  atomicAdd(&pooled[bg * HC + c], h[t]);
  if (c == 0) atomicAdd(&cnt[bg], 1.0f);
}

// ---- final FC ----
__global__ void k_fc(const float* __restrict__ pooled, const float* __restrict__ cnt,
                     const float* __restrict__ W, const float* __restrict__ b,
                     float* __restrict__ out) {
  int t = blockIdx.x * blockDim.x + threadIdx.x;
  if (t >= G_GRAPHS * OUT_C) return;
  int g = t / OUT_C, o = t % OUT_C;
  float cn = cnt[g]; cn = cn > 1.f ? cn : 1.f;
  float s = b[o];
  #pragma unroll
  for (int c = 0; c < HC; ++c) s += (pooled[g * HC + c] / cn) * W[c * OUT_C + o];
  out[t] = s;
}

extern "C" void kernel_launch(void* const* d_in, const int* in_sizes, int n_in,
                              void* d_out, int out_size, void* d_ws, size_t ws_size,
                              hipStream_t stream) {
  (void)in_sizes; (void)n_in; (void)out_size; (void)ws_size;
  const float* x        = (const float*)d_in[0];
  const int*   eidx     = (const int*)d_in[1];
  const float* eattr    = (const float*)d_in[2];
  const int*   batch    = (const int*)d_in[3];
  const float* efc_W    = (const float*)d_in[4];
  const float* efc_b    = (const float*)d_in[5];
  const float* Ws       = (const float*)d_in[6];
  const float* att_src  = (const float*)d_in[7];
  const float* att_dst  = (const float*)d_in[8];
  const float* att_edge = (const float*)d_in[9];
  const float* lin_e    = (const float*)d_in[10];
  const float* biases   = (const float*)d_in[11];
  const float* bn_g     = (const float*)d_in[12];
  const float* bn_b     = (const float*)d_in[13];
  const float* fc_W     = (const float*)d_in[14];
  const float* fc_b     = (const float*)d_in[15];
  const int* srcv = eidx;
  const int* dstv = eidx + N_EDGES;

  char* p = (char*)d_ws;
  auto alloc = [&](size_t bytes) {
    void* r = (void*)p;
    p += (bytes + 255) & ~(size_t)255;
    return r;
  };
  float* WT            = (float*)alloc((size_t)3 * 128 * 128 * 4);
  float* weff          = (float*)alloc((size_t)3 * HEADS * HID * 4);
  float* hcur          = (float*)alloc((size_t)N_NODES * HC * 4);
  float* xl            = (float*)alloc((size_t)N_NODES * HC * 4);
  float* hnx           = (float*)alloc((size_t)N_NODES * HC * 4);
  float* aeall         = (float*)alloc((size_t)3 * N_EDGES * HEADS * 4);
  float* abuf          = (float*)alloc((size_t)N_EDGES * HEADS * 4);
  float* asb           = (float*)alloc((size_t)N_NODES * HEADS * 4);
  float* adb           = (float*)alloc((size_t)N_NODES * HEADS * 4);
  unsigned* amax       = (unsigned*)alloc((size_t)N_NODES * HEADS * 4);
  float* denom         = (float*)alloc((size_t)N_NODES * HEADS * 4);
  float* bnsum         = (float*)alloc(HC * 4);
  float* bnsq          = (float*)alloc(HC * 4);
  float* pooled        = (float*)alloc((size_t)G_GRAPHS * HC * 4);
  float* cnt           = (float*)alloc(G_GRAPHS * 4);

  k_prep_w<<<(3 * 128 * 128 + 255) / 256, 256, 0, stream>>>(Ws, WT);
  k_weff<<<2, 256, 0, stream>>>(lin_e, att_edge, weff);
  k_edge_feat<<<(N_EDGES + 255) / 256, 256, 0, stream>>>(eattr, efc_W, efc_b, weff, aeall);

  for (int l = 0; l < 3; ++l) {
    const float* hin = (l == 0) ? x : hcur;
    k_gemm<<<N_NODES / 16, 256, 0, stream>>>(hin, WT + l * 128 * 128, xl);
    hipMemsetAsync(hnx,   0, (size_t)N_NODES * HC * 4, stream);
    hipMemsetAsync(denom, 0, (size_t)N_NODES * HEADS * 4, stream);
    hipMemsetAsync(amax,  0, (size_t)N_NODES * HEADS * 4, stream);
    hipMemsetAsync(bnsum, 0, HC * 4, stream);
    hipMemsetAsync(bnsq,  0, HC * 4, stream);
    k_attn_node<<<(N_NODES * HEADS + 255) / 256, 256, 0, stream>>>(
        xl, att_src + l * HC, att_dst + l * HC, asb, adb);
    k_edge_a<<<(N_EDGES * HEADS + 255) / 256, 256, 0, stream>>>(
        srcv, dstv, asb, adb, aeall + (size_t)l * N_EDGES * HEADS, abuf, amax);
    k_edge_b<<<(N_EDGES * HC) / 256, 256, 0, stream>>>(
        srcv, dstv, abuf, amax, xl, hnx, denom);
    k_finish_bn<<<(N_NODES + 63) / 64, 128, 0, stream>>>(hnx, denom, biases + l * HC, bnsum, bnsq);
    k_bn_apply<<<(N_NODES * HC + 255) / 256, 256, 0, stream>>>(
        hnx, bnsum, bnsq, bn_g + l * HC, bn_b + l * HC, hcur);
  }

  hipMemsetAsync(pooled, 0, (size_t)G_GRAPHS * HC * 4, stream);
  hipMemsetAsync(cnt,    0, G_GRAPHS * 4, stream);
  k_pool<<<(N_NODES * HC + 255) / 256, 256, 0, stream>>>(hcur, batch, pooled, cnt);
  k_fc<<<2, 256, 0, stream>>>(pooled, cnt, fc_W, fc_b, (float*)d_out);
}